// OrderPooling_42820823941542
// MI455X (gfx1250) — compile-verified
//
#include <hip/hip_runtime.h>

typedef __attribute__((ext_vector_type(2))) float v2f;
typedef __attribute__((ext_vector_type(8))) float v8f;

#define NCLASS 64
#define KGROUP 16
#define XCOLS 256            // NCLASS * (DEPTH+1)
#define XPAD  (XCOLS + 4)    // LDS row padding: bank = (4*M + col) % 64, conflict-free

// Block: 128 threads = 4 wave32. Each block produces out[m0:m0+16, 0:64].
// Phase 1: gather + mean-pool the [16 x 256] x-tile into LDS.
// Phase 2: wave w computes the 16x16 tile at columns [16w, 16w+16) with
//          64 chained v_wmma_f32_16x16x4_f32 (K = 256).
__global__ __launch_bounds__(128)
void order_pool_wmma(const float* __restrict__ h,
                     const int*   __restrict__ idx0,
                     const int*   __restrict__ idx1,
                     const int*   __restrict__ idx2,
                     const int*   __restrict__ idx3,
                     const float* __restrict__ W,
                     const float* __restrict__ bias,
                     float*       __restrict__ out)
{
    __shared__ float xs[16][XPAD];   // ~16.6 KB of the 320 KB WGP LDS

    const int lane = threadIdx.x & 31;
    const int wave = threadIdx.x >> 5;
    const int m0   = blockIdx.x * 16;

    // ---------------- Phase 1: gather + pool (4 rows per wave) ----------------
    for (int r = wave; r < 16; r += 4) {
        const int b = m0 + r;

        // hop 0: plain gather of the center node's 64-float embedding
        {
            const float* src = h + (size_t)idx0[b] * NCLASS;   // wave-uniform base
            v2f v = *(const v2f*)(src + 2 * lane);             // global_load_b64
            *(v2f*)(&xs[r][2 * lane]) = v;
        }

        // hops 1..3: mean over 16 gathered rows each
        const int* ip1 = idx1 + (size_t)b * KGROUP;
        const int* ip2 = idx2 + (size_t)b * KGROUP;
        const int* ip3 = idx3 + (size_t)b * KGROUP;
        const int* hops[3] = { ip1, ip2, ip3 };

        #pragma unroll
        for (int hop = 0; hop < 3; ++hop) {
            const int* ip = hops[hop];
            float a0 = 0.f, a1 = 0.f;
            #pragma unroll
            for (int k = 0; k < KGROUP; ++k) {
                const float* src = h + (size_t)ip[k] * NCLASS; // wave-uniform base
                v2f v = *(const v2f*)(src + 2 * lane);
                a0 += v.x;
                a1 += v.y;
            }
            const float s = 1.0f / (float)KGROUP;
            v2f o; o.x = a0 * s; o.y = a1 * s;
            *(v2f*)(&xs[r][(hop + 1) * NCLASS + 2 * lane]) = o;
        }
    }
    __syncthreads();

    // ---------------- Phase 2: WMMA GEMM out = x @ W.T + b --------------------
    // A (16x4 f32) fragment layout: lanes 0-15 -> M=lane, K={k0,k0+1};
    //                               lanes 16-31 -> M=lane-16, K={k0+2,k0+3}.
    // B (4x16 f32) fragment layout mirrors A with N in place of M.
    const int n0    = wave * 16;
    const int mA    = lane & 15;           // M for A, N for B
    const int kHalf = (lane >> 4) * 2;     // 0 or 2

    v8f acc = {};
    #pragma unroll 4
    for (int k0 = 0; k0 < XCOLS; k0 += 4) {
        v2f a, bm;
        a.x = xs[mA][k0 + kHalf + 0];
        a.y = xs[mA][k0 + kHalf + 1];
        const float* wrow = W + (size_t)(n0 + mA) * XCOLS + k0 + kHalf; // B[k][n] = W[n][k]
        bm.x = wrow[0];
        bm.y = wrow[1];
        // 8 args: (neg_a, A, neg_b, B, c_mod, C, reuse_a, reuse_b)
        acc = __builtin_amdgcn_wmma_f32_16x16x4_f32(false, a, false, bm,
                                                    (short)0, acc, false, false);
    }

    // C/D layout: VGPR r holds M = r + 8*(lane>=16), N = lane&15
    const int   nOut  = n0 + (lane & 15);
    const float bv    = bias[nOut];
    const int   mBase = m0 + ((lane >> 4) << 3);
    #pragma unroll
    for (int r = 0; r < 8; ++r) {
        out[(size_t)(mBase + r) * NCLASS + nOut] = acc[r] + bv;
    }
}

extern "C" void kernel_launch(void* const* d_in, const int* in_sizes, int n_in,
                              void* d_out, int out_size, void* d_ws, size_t ws_size,
                              hipStream_t stream) {
    const float* h    = (const float*)d_in[0];
    const int*   idx0 = (const int*)  d_in[1];
    const int*   idx1 = (const int*)  d_in[2];
    const int*   idx2 = (const int*)  d_in[3];
    const int*   idx3 = (const int*)  d_in[4];
    const float* W    = (const float*)d_in[5];
    const float* b    = (const float*)d_in[6];
    float*       out  = (float*)      d_out;

    const int B = in_sizes[1];          // 65536 batch rows
    const int blocks = B / 16;          // 16 rows per block

    order_pool_wmma<<<blocks, 128, 0, stream>>>(h, idx0, idx1, idx2, idx3, W, b, out);
}